// PaiAutoencoder_66932770341175
// MI455X (gfx1250) — compile-verified
//
#include <hip/hip_runtime.h>
#include <math.h>

// ---------------------------------------------------------------------------
// CDNA5 / gfx1250 fp32 WMMA autoencoder (FeaSt mesh AE).
// ~17 GFLOP / ~120MB model: fully L2-resident (192MB), so fp32 is free.
// All dense contractions run through V_WMMA_F32_16X16X4_F32. Each wave owns a
// 16Mx32N C macro-tile (2 accumulators, shared A fragment) so the K loop is
// pure {2 loads A, 4 loads B, 2 wmma} with zero exec-mask juggling.
// ---------------------------------------------------------------------------

typedef __attribute__((ext_vector_type(2))) float v2f;
typedef __attribute__((ext_vector_type(8))) float v8f;

#define FEAST_HEADS 9

__device__ __forceinline__ float elu_act(float x) {
    return x > 0.0f ? x : (expf(x) - 1.0f);
}

__device__ __forceinline__ v8f wmma4(v2f a, v2f b, v8f c) {
    return __builtin_amdgcn_wmma_f32_16x16x4_f32(false, a, false, b,
                                                 (short)0, c, false, false);
}

// ---------------------------------------------------------------------------
// Batched GEMM:  C[z] = A[z] (MxK, row-major) @ B[z] (KxN, row-major) + bias.
// One wave -> 16x32 C tile. Fragment layouts per ISA 7.12.2:
//   A frag (v2f): lane L -> M = L%16, K = kk + 2*(L>>4) + {0,1}
//   B frag (v2f): lane L -> N = L%16, K = kk + 2*(L>>4) + {0,1}
//   C/D  (v8f):   acc[r] -> M = r + 8*(L>>4), N = L%16
// Out-of-range rows/cols are CLAMPED for loads and masked only at the store,
// so the main K loop has no branches.
// ---------------------------------------------------------------------------
__global__ void gemm16_wmma_f32(const float* __restrict__ A,
                                const float* __restrict__ B,
                                const float* __restrict__ bias,
                                float* __restrict__ C,
                                int M, int N, int K,
                                long long sA, long long sB, long long sC) {
    const int z = blockIdx.z;
    A += (long long)z * sA;
    B += (long long)z * sB;
    C += (long long)z * sC;

    const int lane = threadIdx.x & 31;
    const int half = lane >> 4;   // K-pair selector
    const int l16  = lane & 15;
    const int m0 = blockIdx.y << 4;
    const int n0 = blockIdx.x << 5;          // 32-wide N macro-tile

    const int mrow = min(m0 + l16, M - 1);
    const int n0i  = n0 + l16;
    const int n1i  = n0i + 16;
    const int n0c  = min(n0i, N - 1);
    const int n1c  = min(n1i, N - 1);

    const float* ap  = A + (long long)mrow * K + (half << 1);
    const float* bp0 = B + n0c;
    const float* bp1 = B + n1c;

    v8f acc0 = {}, acc1 = {};
    long long boff = (long long)(half << 1) * N;
    const long long bstep = 4LL * N;
    const int Kmain = K & ~3;

#pragma unroll 2
    for (int kk = 0; kk < Kmain; kk += 4) {
        if ((kk & 127) == 0 && kk + 128 < K) {
            // speculative L2 prefetch (global_prefetch_b8); invalid
            // translations are silently dropped, so no guards needed.
            __builtin_prefetch(ap + kk + 128);
            __builtin_prefetch(bp0 + boff + 128LL * N);
        }
        v2f a, b0, b1;
        a.x  = ap[kk];
        a.y  = ap[kk + 1];
        b0.x = bp0[boff];
        b0.y = bp0[boff + N];
        b1.x = bp1[boff];
        b1.y = bp1[boff + N];
        acc0 = wmma4(a, b0, acc0);
        acc1 = wmma4(a, b1, acc1);
        boff += bstep;
    }
    if (Kmain < K) {  // guarded K tail (single iteration)
        const int kb = Kmain + (half << 1);
        v2f a, b0, b1;
        a.x  = (kb     < K) ? ap[Kmain]        : 0.0f;
        a.y  = (kb + 1 < K) ? ap[Kmain + 1]    : 0.0f;
        b0.x = (kb     < K) ? bp0[boff]        : 0.0f;
        b0.y = (kb + 1 < K) ? bp0[boff + N]    : 0.0f;
        b1.x = (kb     < K) ? bp1[boff]        : 0.0f;
        b1.y = (kb + 1 < K) ? bp1[boff + N]    : 0.0f;
        acc0 = wmma4(a, b0, acc0);
        acc1 = wmma4(a, b1, acc1);
    }

    const float bv0 = bias ? bias[n0c] : 0.0f;
    const float bv1 = bias ? bias[n1c] : 0.0f;
#pragma unroll
    for (int r = 0; r < 8; ++r) {
        const int m = m0 + r + (half << 3);
        if (m < M) {
            float* crow = C + (long long)m * N;
            if (n0i < N) crow[n0i] = acc0[r] + bv0;
            if (n1i < N) crow[n1i] = acc1[r] + bv1;
        }
    }
}

// ---------------------------------------------------------------------------
// Fused FeaSt convolution. One wave = 16 points of one batch element.
// Phase 0: stage mlp_w (F*9 f32) into LDS.
// Phase 1 (lanes 0..15, one point/lane): gather 9 neighbors, head logits =
//   (xn - xn0) @ mlp_w + mlp_b, softmax over heads -> q[9][9], then
//   T[p, f*9 + h] = sum_k xn[k,f] * q[k,h]  -> LDS.
//   NOTE: WMMA K-dim is ordered k' = f*9 + h (the GEMM sum is order-
//   independent) which makes the B operand out_w[k'*O + n] a LINEAR walk.
// Phase 2 (WMMA): out(16 x O) = T(16 x 9F) @ out_w(9F x O), 32-wide N tile,
//   epilogue bias + last-point mask + ELU/identity.
// ---------------------------------------------------------------------------
__global__ void feast_conv_wmma(const float* __restrict__ x,
                                const int* __restrict__ nb,
                                const float* __restrict__ mlp_w,
                                const float* __restrict__ mlp_b,
                                const float* __restrict__ out_w,
                                const float* __restrict__ bias,
                                float* __restrict__ out,
                                int P, int F, int O, int do_elu) {
    extern __shared__ float Tsh[];  // [16 x (9F+1)] T rows, then [9F] mlp_w
    const int b    = blockIdx.y;
    const int p0   = blockIdx.x << 4;
    const int lane = threadIdx.x & 31;
    const int KK   = FEAST_HEADS * F;
    const int tstride = KK + 1;
    float* wl = Tsh + 16 * tstride;

    const float* xb = x + (long long)b * P * F;

    // ---------------- phase 0: stage mlp_w into LDS ----------------
    for (int i = lane; i < KK; i += 32) wl[i] = mlp_w[i];
    __syncthreads();

    // ---------------- phase 1: build T in LDS ----------------
    if (lane < 16) {
        float* trow = Tsh + lane * tstride;
        const int p = p0 + lane;
        if (p < P) {
            int nbr[FEAST_HEADS];
#pragma unroll
            for (int k = 0; k < FEAST_HEADS; ++k) nbr[k] = nb[p * FEAST_HEADS + k];
            const float* x0 = xb + (long long)nbr[0] * F;

            float q[FEAST_HEADS * FEAST_HEADS];
            for (int k = 0; k < FEAST_HEADS; ++k) {
                float lg[FEAST_HEADS];
#pragma unroll
                for (int h = 0; h < FEAST_HEADS; ++h) lg[h] = mlp_b[h];
                const float* xk = xb + (long long)nbr[k] * F;
                for (int f = 0; f < F; ++f) {
                    const float d = xk[f] - x0[f];
#pragma unroll
                    for (int h = 0; h < FEAST_HEADS; ++h)
                        lg[h] += d * wl[f * FEAST_HEADS + h];
                }
                float mx = lg[0];
#pragma unroll
                for (int h = 1; h < FEAST_HEADS; ++h) mx = fmaxf(mx, lg[h]);
                float s = 0.0f;
#pragma unroll
                for (int h = 0; h < FEAST_HEADS; ++h) {
                    const float e = expf(lg[h] - mx);
                    q[k * FEAST_HEADS + h] = e;
                    s += e;
                }
                const float inv = 1.0f / s;
#pragma unroll
                for (int h = 0; h < FEAST_HEADS; ++h) q[k * FEAST_HEADS + h] *= inv;
            }

            for (int f = 0; f < F; ++f) {
                float xv[FEAST_HEADS];
#pragma unroll
                for (int k = 0; k < FEAST_HEADS; ++k)
                    xv[k] = xb[(long long)nbr[k] * F + f];
#pragma unroll
                for (int h = 0; h < FEAST_HEADS; ++h) {
                    float s = 0.0f;
#pragma unroll
                    for (int k = 0; k < FEAST_HEADS; ++k)
                        s += xv[k] * q[k * FEAST_HEADS + h];
                    trow[f * FEAST_HEADS + h] = s;   // K-order = f*9 + h
                }
            }
        } else {
            for (int k = 0; k < KK; ++k) trow[k] = 0.0f;
        }
    }
    __syncthreads();

    // ---------------- phase 2: WMMA projection ----------------
    const int half = lane >> 4;
    const int l16  = lane & 15;
    const float* tp = Tsh + l16 * tstride + (half << 1);
    const int KM = KK & ~3;

    for (int o0 = 0; o0 < O; o0 += 32) {
        const int n0i = o0 + l16;
        const int n1i = n0i + 16;
        const int n0c = min(n0i, O - 1);
        const int n1c = min(n1i, O - 1);
        const float* bp0 = out_w + n0c;
        const float* bp1 = out_w + n1c;

        v8f acc0 = {}, acc1 = {};
        long long boff = (long long)(half << 1) * O;
        const long long bstep = 4LL * O;

#pragma unroll 2
        for (int kk = 0; kk < KM; kk += 4) {
            v2f a, b0, b1;
            a.x  = tp[kk];
            a.y  = tp[kk + 1];
            b0.x = bp0[boff];
            b0.y = bp0[boff + O];
            b1.x = bp1[boff];
            b1.y = bp1[boff + O];
            acc0 = wmma4(a, b0, acc0);
            acc1 = wmma4(a, b1, acc1);
            boff += bstep;
        }
        if (KM < KK) {  // guarded K tail (only F=3: KK=27)
            const int kb = KM + (half << 1);
            v2f a, b0, b1;
            a.x  = (kb     < KK) ? tp[KM]         : 0.0f;
            a.y  = (kb + 1 < KK) ? tp[KM + 1]     : 0.0f;
            b0.x = (kb     < KK) ? bp0[boff]      : 0.0f;
            b0.y = (kb + 1 < KK) ? bp0[boff + O]  : 0.0f;
            b1.x = (kb     < KK) ? bp1[boff]      : 0.0f;
            b1.y = (kb + 1 < KK) ? bp1[boff + O]  : 0.0f;
            acc0 = wmma4(a, b0, acc0);
            acc1 = wmma4(a, b1, acc1);
        }

        const float bv0 = bias[n0c];
        const float bv1 = bias[n1c];
#pragma unroll
        for (int r = 0; r < 8; ++r) {
            const int pl = r + (half << 3);
            const int p  = p0 + pl;
            if (p < P) {
                float* orow = out + ((long long)b * P + p) * O;
                if (n0i < O) {
                    float v = acc0[r] + bv0;
                    if (p == P - 1) v = 0.0f;   // virtual-node mask
                    if (do_elu) v = elu_act(v);
                    orow[n0i] = v;
                }
                if (n1i < O) {
                    float v = acc1[r] + bv1;
                    if (p == P - 1) v = 0.0f;
                    if (do_elu) v = elu_act(v);
                    orow[n1i] = v;
                }
            }
        }
    }
}

// ---------------------------------------------------------------------------
// Host-side launch helpers
// ---------------------------------------------------------------------------
static void launch_feast(const float* xin, const int* nb, const float* mw,
                         const float* mb, const float* ow, const float* bi,
                         float* outp, int P, int F, int O, int elu,
                         hipStream_t s) {
    dim3 g((unsigned)((P + 15) / 16), 16, 1), blk(32, 1, 1);
    size_t sm = (size_t)(16 * (9 * F + 1) + 9 * F) * sizeof(float);
    feast_conv_wmma<<<g, blk, sm, s>>>(xin, nb, mw, mb, ow, bi, outp, P, F, O, elu);
}

static void launch_gemm(const float* A, const float* B, const float* bias,
                        float* C, int M, int N, int K, long long sA,
                        long long sB, long long sC, int batches,
                        hipStream_t s) {
    dim3 g((unsigned)((N + 31) / 32), (unsigned)((M + 15) / 16), (unsigned)batches);
    dim3 blk(32, 1, 1);
    gemm16_wmma_f32<<<g, blk, 0, s>>>(A, B, bias, C, M, N, K, sA, sB, sC);
}

// ---------------------------------------------------------------------------
// kernel_launch: full autoencoder forward pass.
// Input order: 0:x, 1..12 enc{mlp_w,mlp_b,out_w,bias}x3,
// 13..28 dec{...}x4, 29..31 D0..D2, 32..34 U0..U2,
// 35..38 fc_enc_w/b fc_dec_w/b, 39..42 neighbors[0..3] (nb3 unused).
// ---------------------------------------------------------------------------
extern "C" void kernel_launch(void* const* d_in, const int* in_sizes, int n_in,
                              void* d_out, int out_size, void* d_ws,
                              size_t ws_size, hipStream_t stream) {
    (void)in_sizes; (void)n_in; (void)out_size; (void)ws_size;

    const int B = 16;
    const int P0 = 5023, P1 = 1256, P2 = 314, P3 = 79;

    const float* x = (const float*)d_in[0];

    const float *e_mw[3], *e_mb[3], *e_ow[3], *e_bi[3];
    const float *c_mw[4], *c_mb[4], *c_ow[4], *c_bi[4];
    int idx = 1;
    for (int i = 0; i < 3; ++i) {
        e_mw[i] = (const float*)d_in[idx++];
        e_mb[i] = (const float*)d_in[idx++];
        e_ow[i] = (const float*)d_in[idx++];
        e_bi[i] = (const float*)d_in[idx++];
    }
    for (int i = 0; i < 4; ++i) {
        c_mw[i] = (const float*)d_in[idx++];
        c_mb[i] = (const float*)d_in[idx++];
        c_ow[i] = (const float*)d_in[idx++];
        c_bi[i] = (const float*)d_in[idx++];
    }
    const float* D0 = (const float*)d_in[idx++];
    const float* D1 = (const float*)d_in[idx++];
    const float* D2 = (const float*)d_in[idx++];
    const float* U0 = (const float*)d_in[idx++];
    const float* U1 = (const float*)d_in[idx++];
    const float* U2 = (const float*)d_in[idx++];
    const float* few = (const float*)d_in[idx++];
    const float* feb = (const float*)d_in[idx++];
    const float* fdw = (const float*)d_in[idx++];
    const float* fdb = (const float*)d_in[idx++];
    const int* nb0 = (const int*)d_in[idx++];
    const int* nb1 = (const int*)d_in[idx++];
    const int* nb2 = (const int*)d_in[idx++];

    float* outp = (float*)d_out;

    // Two ping-pong activation regions (max activation: 16 x 5023 x 32 f32).
    const long long MAXBUF = (long long)B * P0 * 32;
    float* bufA = (float*)d_ws;
    float* bufB = bufA + MAXBUF;

    // -------- encoder --------
    launch_feast(x, nb0, e_mw[0], e_mb[0], e_ow[0], e_bi[0], bufA,
                 P0, 3, 32, 1, stream);                               // conv0
    launch_gemm(D0, bufA, nullptr, bufB, P1, 32, P0,
                0, (long long)P0 * 32, (long long)P1 * 32, B, stream); // pool D0
    launch_feast(bufB, nb1, e_mw[1], e_mb[1], e_ow[1], e_bi[1], bufA,
                 P1, 32, 64, 1, stream);                              // conv1
    launch_gemm(D1, bufA, nullptr, bufB, P2, 64, P1,
                0, (long long)P1 * 64, (long long)P2 * 64, B, stream); // pool D1
    launch_feast(bufB, nb2, e_mw[2], e_mb[2], e_ow[2], e_bi[2], bufA,
                 P2, 64, 128, 1, stream);                             // conv2
    launch_gemm(D2, bufA, nullptr, bufB, P3, 128, P2,
                0, (long long)P2 * 128, (long long)P3 * 128, B, stream); // pool D2

    // -------- bottleneck FCs --------
    const int FLAT = P3 * 128;  // 10112
    launch_gemm(bufB, few, feb, bufA, B, 256, FLAT, 0, 0, 0, 1, stream);  // fc_enc
    launch_gemm(bufA, fdw, fdb, bufB, B, FLAT, 256, 0, 0, 0, 1, stream);  // fc_dec

    // -------- decoder --------
    launch_gemm(U2, bufB, nullptr, bufA, P2, 128, P3,
                0, (long long)P3 * 128, (long long)P2 * 128, B, stream); // unpool U2
    launch_feast(bufA, nb2, c_mw[0], c_mb[0], c_ow[0], c_bi[0], bufB,
                 P2, 128, 64, 1, stream);                             // dec conv0
    launch_gemm(U1, bufB, nullptr, bufA, P1, 64, P2,
                0, (long long)P2 * 64, (long long)P1 * 64, B, stream); // unpool U1
    launch_feast(bufA, nb1, c_mw[1], c_mb[1], c_ow[1], c_bi[1], bufB,
                 P1, 64, 32, 1, stream);                              // dec conv1
    launch_gemm(U0, bufB, nullptr, bufA, P0, 32, P1,
                0, (long long)P1 * 32, (long long)P0 * 32, B, stream); // unpool U0
    launch_feast(bufA, nb0, c_mw[2], c_mb[2], c_ow[2], c_bi[2], bufB,
                 P0, 32, 32, 1, stream);                              // dec conv2
    launch_feast(bufB, nb0, c_mw[3], c_mb[3], c_ow[3], c_bi[3], outp,
                 P0, 32, 3, 0, stream);                               // dec conv3
}